// DeMash_2671469658782
// MI455X (gfx1250) — compile-verified
//
#include <hip/hip_runtime.h>

// Problem constants (from reference): B=2, R=1, A=16, S=14, F=1024, N=512
#define DD      7168          // D = S*N, GEMM output-col and K dim
#define SDIM    14
#define FDIM    1024
#define ADIM    16
#define BDIM    2
#define NSC     512
#define MROWS   32            // B*R*A
#define KPAIRS  896           // DD / 8   (two WMMA K-steps per pair)
#define SPLITK  8
#define KPW     (KPAIRS / SPLITK)    // 112 k-pairs per wave
#define ETILES  (DD / 32)     // 224 (32 e-rows per wave)
#define OUT_N   917504        // 2*B*R*A*S*F floats
#define ACC_N   (2 * DD * MROWS)       // 458752 floats (re plane + im plane)
#define PACK_N  (KPAIRS * 2 * 2 * 128) // 458752 floats

typedef float v2f __attribute__((ext_vector_type(2)));
typedef float v4f __attribute__((ext_vector_type(4)));
typedef float v8f __attribute__((ext_vector_type(8)));

// ---------------------------------------------------------------------------
// Kernel 0: zero output grid and the split-K accumulator planes
// ---------------------------------------------------------------------------
__global__ __launch_bounds__(256) void dm_zero(float* __restrict__ out,
                                               float* __restrict__ acc) {
    int tid = blockIdx.x * 256 + threadIdx.x;
    if (tid < OUT_N) out[tid] = 0.0f;
    if (tid < ACC_N) acc[tid] = 0.0f;
}

// ---------------------------------------------------------------------------
// Kernel 1: gather x by sc_ind and pack X^T into WMMA B layout, float4/lane.
// P[kk][p][h][lane] = float4{ X[p][m][db], X[p][m][db+1], X[p][m][db+2],
//                             X[p][m][db+3] },  db = 8*kk + 4*(lane>>4),
// m = h*16 + (lane&15).  Step0 of a pair uses (.xy), step1 uses (.zw):
// K-set step0 = {d0,d0+1,d0+4,d0+5}, step1 = {d0+2,d0+3,d0+6,d0+7}.
// ---------------------------------------------------------------------------
__global__ __launch_bounds__(256) void dm_pack(const float* __restrict__ x_re,
                                               const float* __restrict__ x_im,
                                               const int*   __restrict__ sc_ind,
                                               float* __restrict__ P) {
    int tid = blockIdx.x * 256 + threadIdx.x;       // 0 .. KPAIRS*2*2*32-1
    if (tid >= KPAIRS * 2 * 2 * 32) return;
    int lane = tid & 31;
    int h    = (tid >> 5) & 1;
    int p    = (tid >> 6) & 1;
    int kk   = tid >> 7;

    int m = h * 16 + (lane & 15);
    int b = m >> 4;
    int a = m & 15;
    int dbase = 8 * kk + ((lane >> 4) << 2);

    const float* src = (p == 0) ? x_re : x_im;
    float v[4];
#pragma unroll
    for (int j = 0; j < 4; ++j) {
        int d   = dbase + j;
        int s   = d >> 9;            // d / 512
        int nsc = d & (NSC - 1);     // d % 512
        int f   = sc_ind[nsc];
        v[j] = src[(((size_t)(b * ADIM + a)) * SDIM + s) * FDIM + f];
    }
    *(v4f*)(P + (size_t)tid * 4) = v4f{v[0], v[1], v[2], v[3]};
}

// ---------------------------------------------------------------------------
// Kernel 2: Yt_r[e,m] = sum_d Cr[e,d]*Xr[m,d] + Ci[e,d]*Xi[m,d]
//           Yt_i[e,m] = sum_d Cr[e,d]*Xi[m,d] - Ci[e,d]*Xr[m,d]
// 32 e-rows per wave (2 row-groups), b128 loads cover 2 WMMA K-steps each.
// Split-K=8 with f32 global atomics. 4 waves / block.
// ---------------------------------------------------------------------------
__global__ __launch_bounds__(128) void dm_gemm(const float* __restrict__ Cre,
                                               const float* __restrict__ Cim,
                                               const float* __restrict__ P,
                                               float* __restrict__ acc) {
    const int lane = threadIdx.x & 31;
    const int gw   = blockIdx.x * 4 + (threadIdx.x >> 5);
    const int etile = gw / SPLITK;
    const int kseg  = gw % SPLITK;
    const int e0    = etile * 32;

    const int row  = lane & 15;          // e row within row-group
    const int dsel = (lane >> 4) << 2;   // 0 or 4 (K half select)

    const float* aLoR = Cre + (size_t)(e0 + row) * DD + dsel;
    const float* aLoI = Cim + (size_t)(e0 + row) * DD + dsel;
    const float* aHiR = Cre + (size_t)(e0 + 16 + row) * DD + dsel;
    const float* aHiI = Cim + (size_t)(e0 + 16 + row) * DD + dsel;

    // group 0: rows e0..e0+15 ; group 1: rows e0+16..e0+31
    v8f r00 = {}, r01 = {}, i00 = {}, i01 = {};
    v8f r10 = {}, r11 = {}, i10 = {}, i11 = {};

    const int kk0 = kseg * KPW;
    const int kk1 = kk0 + KPW;
    for (int kk = kk0; kk < kk1; ++kk) {
        // A operands: one NT b128 per lane per (group, part) covers 2 K-steps
        v4f aLr = __builtin_nontemporal_load((const v4f*)(aLoR + 8 * (size_t)kk));
        v4f aLi = __builtin_nontemporal_load((const v4f*)(aLoI + 8 * (size_t)kk));
        v4f aHr = __builtin_nontemporal_load((const v4f*)(aHiR + 8 * (size_t)kk));
        v4f aHi = __builtin_nontemporal_load((const v4f*)(aHiI + 8 * (size_t)kk));
        // B operands: coalesced 512B/wave b128 loads from packed buffer
        const float* pb = P + (size_t)kk * 512 + lane * 4;
        v4f br0 = *(const v4f*)(pb + 0);     // p=re, m 0-15
        v4f br1 = *(const v4f*)(pb + 128);   // p=re, m 16-31
        v4f bi0 = *(const v4f*)(pb + 256);   // p=im, m 0-15
        v4f bi1 = *(const v4f*)(pb + 384);   // p=im, m 16-31

#pragma unroll
        for (int st = 0; st < 2; ++st) {
            v2f ALr = v2f{aLr[2*st], aLr[2*st+1]};
            v2f ALi = v2f{aLi[2*st], aLi[2*st+1]};
            v2f AHr = v2f{aHr[2*st], aHr[2*st+1]};
            v2f AHi = v2f{aHi[2*st], aHi[2*st+1]};
            v2f Br0 = v2f{br0[2*st], br0[2*st+1]};
            v2f Br1 = v2f{br1[2*st], br1[2*st+1]};
            v2f Bi0 = v2f{bi0[2*st], bi0[2*st+1]};
            v2f Bi1 = v2f{bi1[2*st], bi1[2*st+1]};
            v2f ALn = v2f{-ALi.x, -ALi.y};   // f32 WMMA lacks A-neg modifier
            v2f AHn = v2f{-AHi.x, -AHi.y};

            r00 = __builtin_amdgcn_wmma_f32_16x16x4_f32(false, ALr, false, Br0, (short)0, r00, false, false);
            r00 = __builtin_amdgcn_wmma_f32_16x16x4_f32(false, ALi, false, Bi0, (short)0, r00, false, false);
            r01 = __builtin_amdgcn_wmma_f32_16x16x4_f32(false, ALr, false, Br1, (short)0, r01, false, false);
            r01 = __builtin_amdgcn_wmma_f32_16x16x4_f32(false, ALi, false, Bi1, (short)0, r01, false, false);
            i00 = __builtin_amdgcn_wmma_f32_16x16x4_f32(false, ALr, false, Bi0, (short)0, i00, false, false);
            i00 = __builtin_amdgcn_wmma_f32_16x16x4_f32(false, ALn, false, Br0, (short)0, i00, false, false);
            i01 = __builtin_amdgcn_wmma_f32_16x16x4_f32(false, ALr, false, Bi1, (short)0, i01, false, false);
            i01 = __builtin_amdgcn_wmma_f32_16x16x4_f32(false, ALn, false, Br1, (short)0, i01, false, false);

            r10 = __builtin_amdgcn_wmma_f32_16x16x4_f32(false, AHr, false, Br0, (short)0, r10, false, false);
            r10 = __builtin_amdgcn_wmma_f32_16x16x4_f32(false, AHi, false, Bi0, (short)0, r10, false, false);
            r11 = __builtin_amdgcn_wmma_f32_16x16x4_f32(false, AHr, false, Br1, (short)0, r11, false, false);
            r11 = __builtin_amdgcn_wmma_f32_16x16x4_f32(false, AHi, false, Bi1, (short)0, r11, false, false);
            i10 = __builtin_amdgcn_wmma_f32_16x16x4_f32(false, AHr, false, Bi0, (short)0, i10, false, false);
            i10 = __builtin_amdgcn_wmma_f32_16x16x4_f32(false, AHn, false, Br0, (short)0, i10, false, false);
            i11 = __builtin_amdgcn_wmma_f32_16x16x4_f32(false, AHr, false, Bi1, (short)0, i11, false, false);
            i11 = __builtin_amdgcn_wmma_f32_16x16x4_f32(false, AHn, false, Br1, (short)0, i11, false, false);
        }
    }

    // Split-K combine: D tile layout -> lane l, vgpr v holds (v + 8*(l>=16), m=l&15)
    float* accR = acc;                       // [DD][32]
    float* accI = acc + (size_t)DD * MROWS;
    const int col    = lane & 15;
    const int rowAdd = (lane >> 4) * 8;
#pragma unroll
    for (int v = 0; v < 8; ++v) {
        size_t eL = (size_t)(e0 + v + rowAdd);
        size_t eH = eL + 16;
        atomicAdd(&accR[eL * MROWS +      col], r00[v]);
        atomicAdd(&accR[eL * MROWS + 16 + col], r01[v]);
        atomicAdd(&accI[eL * MROWS +      col], i00[v]);
        atomicAdd(&accI[eL * MROWS + 16 + col], i01[v]);
        atomicAdd(&accR[eH * MROWS +      col], r10[v]);
        atomicAdd(&accR[eH * MROWS + 16 + col], r11[v]);
        atomicAdd(&accI[eH * MROWS +      col], i10[v]);
        atomicAdd(&accI[eH * MROWS + 16 + col], i11[v]);
    }
}

// ---------------------------------------------------------------------------
// Kernel 3: scatter acc planes into out[2,B,R,A,S,F] at f = sc_ind[nsc]
// ---------------------------------------------------------------------------
__global__ __launch_bounds__(256) void dm_scatter(const float* __restrict__ acc,
                                                  const int*   __restrict__ sc_ind,
                                                  float* __restrict__ out) {
    int tid = blockIdx.x * 256 + threadIdx.x;   // 0 .. 2*B*A*S*NSC-1
    if (tid >= 2 * BDIM * ADIM * SDIM * NSC) return;
    int nsc  = tid & (NSC - 1);
    int t    = tid >> 9;
    int s    = t % SDIM;  t /= SDIM;
    int a    = t % ADIM;  t /= ADIM;
    int b    = t % BDIM;  t /= BDIM;
    int part = t;                                // 0 = real, 1 = imag

    int f = sc_ind[nsc];
    size_t e = (size_t)s * NSC + nsc;
    int m = b * ADIM + a;
    float v = acc[(size_t)part * DD * MROWS + e * MROWS + m];
    size_t oidx = ((((size_t)(part * BDIM + b)) * ADIM + a) * SDIM + s) * FDIM + f;
    out[oidx] = v;
}

// ---------------------------------------------------------------------------
extern "C" void kernel_launch(void* const* d_in, const int* in_sizes, int n_in,
                              void* d_out, int out_size, void* d_ws, size_t ws_size,
                              hipStream_t stream) {
    const float* x_re   = (const float*)d_in[0];
    const float* x_im   = (const float*)d_in[1];
    const float* C_re   = (const float*)d_in[2];
    const float* C_im   = (const float*)d_in[3];
    const int*   sc_ind = (const int*)d_in[4];
    float* out = (float*)d_out;

    float* P   = (float*)d_ws;          // PACK_N floats (1.75 MB)
    float* acc = P + PACK_N;            // ACC_N floats (1.75 MB)

    // 0) zero output + accumulator
    dm_zero<<<(OUT_N + 255) / 256, 256, 0, stream>>>(out, acc);
    // 1) gather + pack X^T into WMMA B layout (float4 per lane)
    dm_pack<<<(KPAIRS * 2 * 2 * 32 + 255) / 256, 256, 0, stream>>>(x_re, x_im, sc_ind, P);
    // 2) streaming f32 WMMA GEMM: 32 e-rows/wave, split-K=8, 4 waves/block
    dm_gemm<<<ETILES * SPLITK / 4, 128, 0, stream>>>(C_re, C_im, P, acc);
    // 3) scatter into padded grid
    dm_scatter<<<(2 * BDIM * ADIM * SDIM * NSC + 255) / 256, 256, 0, stream>>>(acc, sc_ind, out);
}